// HierLrGreen_32719060861299
// MI455X (gfx1250) — compile-verified
//
#include <hip/hip_runtime.h>

// CDNA5 / gfx1250, wave32. Butterfly low-rank operator:
//   out[b,o,k] = sum_l sum_r U1_l[o,k,r] * ( sum_i U2_l[i,k,r] * x[b,i,k^mask_l] )
// Per (mode,level): two tiny GEMMs on V_WMMA_F32_16X16X4_F32.
//   stage1: D1[r,b] = U2^T(16x4-of-64) x x'(64x16)   (16 K-steps)
//   stage2: out[b,o] += t(16x4) x U1^T(4x16)         (4 fully-used WMMAs)

typedef __attribute__((ext_vector_type(2))) float v2f;
typedef __attribute__((ext_vector_type(4))) float v4f;
typedef __attribute__((ext_vector_type(8))) float v8f;
typedef __attribute__((ext_vector_type(4))) int   v4i;

#define MODES 8192
#define NCH   64
#define RANK  4
#define NLEV  10
#define NB    16
#define MT    16   // modes per workgroup
#define MPW   2    // modes per wave (8 waves / block)
#define OPITCH 17  // padded LDS pitch for output transpose tile

#if __has_builtin(__builtin_amdgcn_global_load_async_to_lds_b128)
#define HAVE_ASYNC_LDS 1
#else
#define HAVE_ASYNC_LDS 0
#endif

struct UPtrs { const float* u1[NLEV]; const float* u2[NLEV]; };

__device__ __forceinline__ float swap16(float v) {
  // lanes 0..15 <-> 16..31 (SWAPX16: and=0x1f, or=0, xor=0x10)
  return __int_as_float(__builtin_amdgcn_ds_swizzle(__float_as_int(v), 0x401f));
}

// ---------------- Pass 1: x[b,i,k] -> xT[k][i][b]  (b fastest, 64B per (k,i) row)
__global__ __launch_bounds__(256) void xpose_kernel(const float* __restrict__ x,
                                                    float* __restrict__ xT) {
  __shared__ float tile[64][17];           // [kLoc][b], padded
  const int i  = blockIdx.y;               // channel
  const int k0 = blockIdx.x * 64;          // mode tile
  const int t  = threadIdx.x;
  {
    const int b  = t >> 4;                 // 0..15
    const int kk = (t & 15) * 4;           // 0..60
    const float4 v = *(const float4*)(x + ((size_t)(b * NCH + i)) * MODES + k0 + kk);
    tile[kk + 0][b] = v.x; tile[kk + 1][b] = v.y;
    tile[kk + 2][b] = v.z; tile[kk + 3][b] = v.w;
  }
  __syncthreads();
  {
    const int kLoc = t >> 2;               // 0..63
    const int b0   = (t & 3) * 4;          // 0,4,8,12
    float4 v;
    v.x = tile[kLoc][b0 + 0]; v.y = tile[kLoc][b0 + 1];
    v.z = tile[kLoc][b0 + 2]; v.w = tile[kLoc][b0 + 3];
    *(float4*)(xT + (((size_t)(k0 + kLoc)) * NCH + i) * NB + b0) = v;
  }
}

// ---------------- Pass 2: main butterfly kernel
__global__ __launch_bounds__(256) void hier_kernel(const float* __restrict__ xT,
                                                   UPtrs up,
                                                   float* __restrict__ out) {
  extern __shared__ float lds[];
  float* u2s  = lds;                       // [64 i][MT m][4 r] = 16 KB
  float* u1s  = lds + NCH * MT * RANK;     // [64 o][MT m][4 r] = 16 KB
  float* outs = lds + 2 * NCH * MT * RANK; // [1024 row][OPITCH] = 68 KB

  const int t    = threadIdx.x;
  const int lane = t & 31;
  const int wave = t >> 5;
  const int hi   = lane >> 4;              // 0 | 1 (half-wave)
  const int ln   = lane & 15;
  const int k0   = blockIdx.x * MT;

  // per-lane offsets (constant for the whole kernel)
  const int xoff  = hi * 32 + ln;          // x'  element (i=2*hi, b=ln), in floats
  const int rsel  = ln & 3;                // A rows 4..15 duplicate rows 0..3 (unused)
  const int aoff0 = 2 * hi * MT * RANK + rsel;     // U2^T lane base (floats)
  const int b2off = ln * MT * RANK + 2 * hi;       // U1^T lane base (floats)

  v8f acc[MPW][4];
#pragma unroll
  for (int mm = 0; mm < MPW; ++mm)
#pragma unroll
    for (int ot = 0; ot < 4; ++ot)
      acc[mm][ot] = (v8f){0.f, 0.f, 0.f, 0.f, 0.f, 0.f, 0.f, 0.f};

  for (int l = 0; l < NLEV; ++l) {
    __syncthreads();
    // ---- stage this level's U factors into LDS as [i][m][r]
    if (l == 0) {
      // diag factors: layout [i][r][MODES]
      for (int e = t; e < NCH * MT; e += 256) {
        const int i_ = e / MT, m = e % MT;
#pragma unroll
        for (int r = 0; r < RANK; ++r) {
          u1s[e * RANK + r] = up.u1[0][((size_t)i_ * RANK + r) * MODES + k0 + m];
          u2s[e * RANK + r] = up.u2[0][((size_t)i_ * RANK + r) * MODES + k0 + m];
        }
      }
    } else {
      // hierarchical factors: layout [i][MODES][RANK] (16B per (i,mode))
      for (int e = t; e < NCH * MT; e += 256) {
        const int i_ = e / MT, m = e % MT;
        const float* g1 = up.u1[l] + ((size_t)i_ * MODES + k0 + m) * RANK;
        const float* g2 = up.u2[l] + ((size_t)i_ * MODES + k0 + m) * RANK;
#if HAVE_ASYNC_LDS
        __builtin_amdgcn_global_load_async_to_lds_b128((v4i*)g1, (v4i*)&u1s[e * RANK], 0, 0);
        __builtin_amdgcn_global_load_async_to_lds_b128((v4i*)g2, (v4i*)&u2s[e * RANK], 0, 0);
#else
        *(v4f*)&u1s[e * RANK] = *(const v4f*)g1;
        *(v4f*)&u2s[e * RANK] = *(const v4f*)g2;
#endif
      }
#if HAVE_ASYNC_LDS
      asm volatile("s_wait_asynccnt 0x0" ::: "memory");
#endif
    }
    __syncthreads();

    const int mask = (l == 0) ? 0 : (1 << (l - 1));

    for (int mm = 0; mm < MPW; ++mm) {
      // wave-uniform scalars: force SGPR so loads use saddr + small voffset
      const int mLoc = __builtin_amdgcn_readfirstlane(wave * MPW + mm);
      const int krow = __builtin_amdgcn_readfirstlane((k0 + mLoc) ^ mask);
      const float* xrow = xT + ((size_t)krow << 10);   // 1024 floats per mode row
      const int aoff = aoff0 + mLoc * RANK;

      // ---- stage 1: D1[r,b] accumulate over K=64 in 16 WMMA steps
      v8f d1a = (v8f){0.f,0.f,0.f,0.f,0.f,0.f,0.f,0.f};
      v8f d1b = (v8f){0.f,0.f,0.f,0.f,0.f,0.f,0.f,0.f};
#pragma unroll
      for (int j = 0; j < 16; ++j) {
        v2f A, B;
        // A (16x4, M=r, K=i): pair (i, i+1) for row r
        A.x = u2s[aoff + j * 4 * MT * RANK];
        A.y = u2s[aoff + j * 4 * MT * RANK + MT * RANK];
        // B (4x16, K=i, N=b): coalesced 64B rows of xT, imm offsets off one base
        B.x = xrow[j * 64 + xoff];
        B.y = xrow[j * 64 + xoff + NB];
        if (j & 1)
          d1b = __builtin_amdgcn_wmma_f32_16x16x4_f32(false, A, false, B, (short)0, d1b, false, false);
        else
          d1a = __builtin_amdgcn_wmma_f32_16x16x4_f32(false, A, false, B, (short)0, d1a, false, false);
      }
      v8f d1;
#pragma unroll
      for (int q = 0; q < 8; ++q) d1[q] = d1a[q] + d1b[q];

      // ---- t[b,r] sits in d1[0..3] lanes 0..15; build A2 (16x4, M=b, K=r)
      const float s2 = swap16(d1[2]);
      const float s3 = swap16(d1[3]);
      v2f A2;
      A2.x = hi ? s2 : d1[0];
      A2.y = hi ? s3 : d1[1];

      // ---- stage 2: out[b, o-tile] += t(16x4) x U1^T(4x16)
#pragma unroll
      for (int ot = 0; ot < 4; ++ot) {
        // 8B-aligned pair (r=2hi, 2hi+1) -> single ds_load_b64
        const v2f B2 = *(const v2f*)&u1s[b2off + mLoc * RANK + ot * 16 * MT * RANK];
        acc[mm][ot] = __builtin_amdgcn_wmma_f32_16x16x4_f32(false, A2, false, B2, (short)0,
                                                            acc[mm][ot], false, false);
      }
    }
  }

  // ---- transpose accumulators through LDS, store coalesced 64B runs
  __syncthreads();
#pragma unroll
  for (int mm = 0; mm < MPW; ++mm) {
    const int mLoc = wave * MPW + mm;
#pragma unroll
    for (int ot = 0; ot < 4; ++ot) {
#pragma unroll
      for (int v = 0; v < 8; ++v) {
        const int b = v + 8 * hi;          // D tile: vgpr v = rows (v, v+8)
        const int o = ot * 16 + ln;
        outs[(b * NCH + o) * OPITCH + mLoc] = acc[mm][ot][v];
      }
    }
  }
  __syncthreads();
  for (int pass = 0; pass < 16; ++pass) {
    const int row = pass * 64 + (t >> 2);  // row = b*64 + o
    const int seg = (t & 3) * 4;           // 4 floats per thread, 64B per 4 threads
    float4 v;
    v.x = outs[row * OPITCH + seg + 0];
    v.y = outs[row * OPITCH + seg + 1];
    v.z = outs[row * OPITCH + seg + 2];
    v.w = outs[row * OPITCH + seg + 3];
    *(float4*)(out + (size_t)row * MODES + k0 + seg) = v;
  }
}

extern "C" void kernel_launch(void* const* d_in, const int* in_sizes, int n_in,
                              void* d_out, int out_size, void* d_ws, size_t ws_size,
                              hipStream_t stream) {
  (void)in_sizes; (void)n_in; (void)out_size; (void)ws_size;
  const float* x = (const float*)d_in[0];
  UPtrs up;
  up.u1[0] = (const float*)d_in[1];   // dU1
  up.u2[0] = (const float*)d_in[2];   // dU2
  for (int l = 1; l < NLEV; ++l) {
    up.u1[l] = (const float*)d_in[2 + l];              // U1_list[l-1]
    up.u2[l] = (const float*)d_in[2 + (NLEV - 1) + l]; // U2_list[l-1]
  }
  float* xT  = (float*)d_ws;          // 8192*64*16 floats = 32 MB scratch
  float* out = (float*)d_out;

  xpose_kernel<<<dim3(MODES / 64, NCH), 256, 0, stream>>>(x, xT);

  const size_t smem = (2 * NCH * MT * RANK + 1024 * OPITCH) * sizeof(float); // ~100 KB
  hier_kernel<<<MODES / MT, 256, smem, stream>>>(xT, up, out);
}